// LongRangeFeaturizer_51788715655672
// MI455X (gfx1250) — compile-verified
//
#include <hip/hip_runtime.h>

typedef __attribute__((ext_vector_type(2))) float v2f;
typedef __attribute__((ext_vector_type(8))) float v8f;

#define SYS 32
#define NA  512
#define FD  256
#define SN  (SYS * NA)

// ---------------------------------------------------------------------------
// Kernel 1: q = features @ W^T + b     [SN, FD] = [SN, FD] x [FD, FD]^T
// grid = SN/16 blocks, 128 threads (4 waves). Wave w computes rows r0..r0+15,
// cols w*64 .. w*64+63 (4 WMMA C tiles, A fragment reused across the 4 tiles).
// ---------------------------------------------------------------------------
__global__ void charges_gemm_k(const float* __restrict__ feat,
                               const float* __restrict__ W,
                               const float* __restrict__ bias,
                               float* __restrict__ q) {
  const int r0   = blockIdx.x * 16;
  const int wave = threadIdx.x >> 5;
  const int lane = threadIdx.x & 31;
  const int half = lane >> 4;     // 0: K={0,1}, 1: K={2,3}
  const int m    = lane & 15;     // M (for A) / N (for B)
  const int c0   = wave * 64;

  v8f acc0 = {}, acc1 = {}, acc2 = {}, acc3 = {};

  const float* arow = feat + (size_t)(r0 + m) * FD;
  // B[k][n] = W^T[k][n] = W[n][k]  -> contiguous K pairs along W rows
  const float* wr0 = W + (size_t)(c0 +  0 + m) * FD;
  const float* wr1 = W + (size_t)(c0 + 16 + m) * FD;
  const float* wr2 = W + (size_t)(c0 + 32 + m) * FD;
  const float* wr3 = W + (size_t)(c0 + 48 + m) * FD;

  for (int k0 = 0; k0 < FD; k0 += 4) {
    const int kb = k0 + 2 * half;
    v2f a  = *(const v2f*)(arow + kb);
    v2f b0 = *(const v2f*)(wr0 + kb);
    v2f b1 = *(const v2f*)(wr1 + kb);
    v2f b2 = *(const v2f*)(wr2 + kb);
    v2f b3 = *(const v2f*)(wr3 + kb);
    acc0 = __builtin_amdgcn_wmma_f32_16x16x4_f32(false, a, false, b0, (short)0, acc0, false, false);
    acc1 = __builtin_amdgcn_wmma_f32_16x16x4_f32(false, a, false, b1, (short)0, acc1, false, false);
    acc2 = __builtin_amdgcn_wmma_f32_16x16x4_f32(false, a, false, b2, (short)0, acc2, false, false);
    acc3 = __builtin_amdgcn_wmma_f32_16x16x4_f32(false, a, false, b3, (short)0, acc3, false, false);
  }

  // C layout: VGPR r -> M = r + 8*half, N = m
  #pragma unroll
  for (int r = 0; r < 8; ++r) {
    const size_t row = (size_t)(r0 + r + 8 * half);
    const int col    = c0 + m;
    q[row * FD + col +  0] = acc0[r] + bias[col +  0];
    q[row * FD + col + 16] = acc1[r] + bias[col + 16];
    q[row * FD + col + 32] = acc2[r] + bias[col + 32];
    q[row * FD + col + 48] = acc3[r] + bias[col + 48];
  }
}

// ---------------------------------------------------------------------------
// Kernel 2: fused  out = (0.5 * v(r) @ q) .* q   per system.
// grid = SYS * (NA/16) blocks, 256 threads (8 waves).
// Block handles (system s, row tile i). Per 64-wide j chunk the block
// cooperatively evaluates the 16x64 pair-potential tile into LDS, then each
// wave runs 16 WMMA K-steps for its 2 column tiles (32 of the 256 features).
// ---------------------------------------------------------------------------
__global__ void potential_gemm_k(const float* __restrict__ pos,
                                 const float* __restrict__ q,
                                 float* __restrict__ out) {
  __shared__ float vtile[16 * 64];   // 4 KB: v[m][j_local]

  const int s     = blockIdx.x >> 5;     // NA/16 == 32 row tiles per system
  const int itile = blockIdx.x & 31;
  const int i16   = itile * 16;
  const int tid   = threadIdx.x;
  const int wave  = tid >> 5;
  const int lane  = tid & 31;
  const int half  = lane >> 4;
  const int m     = lane & 15;
  const int n0    = wave * 32;           // 8 waves x 32 cols = 256 features

  // cooperative v-tile work assignment: 16 threads per row, 4 j's each
  const int vm  = tid & 15;
  const int vj  = (tid >> 4) * 4;
  const float* ps = pos + (size_t)s * NA * 3;
  const int rowAtom = i16 + vm;
  const float rx = ps[rowAtom * 3 + 0];
  const float ry = ps[rowAtom * 3 + 1];
  const float rz = ps[rowAtom * 3 + 2];

  const float* qs = q + (size_t)s * NA * FD;
  const float PI_over_R = 3.14159265358979323846f / 5.0f;

  v8f acc0 = {}, acc1 = {};

  for (int jc = 0; jc < NA; jc += 64) {
    // ---- stage v[16][64] into LDS ----
    #pragma unroll
    for (int c = 0; c < 4; ++c) {
      const int j = jc + vj + c;
      const float dx = rx - ps[j * 3 + 0];
      const float dy = ry - ps[j * 3 + 1];
      const float dz = rz - ps[j * 3 + 2];
      const float d  = sqrtf(dx * dx + dy * dy + dz * dz);
      float v = 0.0f;
      if (rowAtom != j) {
        const float fcut = (d < 5.0f) ? (0.5f + 0.5f * __cosf(PI_over_R * d)) : 0.0f;
        v = (1.0f - fcut) / d;
      }
      vtile[vm * 64 + vj + c] = v;
    }
    __syncthreads();

    // ---- 16 K-steps of WMMA over this chunk ----
    #pragma unroll
    for (int kk = 0; kk < 64; kk += 4) {
      const int kb = kk + 2 * half;                  // K pair for this lane half
      v2f a = *(const v2f*)(&vtile[m * 64 + kb]);    // ds_load_b64
      const size_t jrow = (size_t)(jc + kb);
      v2f b0, b1;
      b0.x = qs[(jrow + 0) * FD + n0 + m];
      b0.y = qs[(jrow + 1) * FD + n0 + m];
      b1.x = qs[(jrow + 0) * FD + n0 + 16 + m];
      b1.y = qs[(jrow + 1) * FD + n0 + 16 + m];
      acc0 = __builtin_amdgcn_wmma_f32_16x16x4_f32(false, a, false, b0, (short)0, acc0, false, false);
      acc1 = __builtin_amdgcn_wmma_f32_16x16x4_f32(false, a, false, b1, (short)0, acc1, false, false);
    }
    __syncthreads();
  }

  // ---- fused epilogue: out = 0.5 * V * q ----
  #pragma unroll
  for (int r = 0; r < 8; ++r) {
    const size_t rowg = (size_t)s * NA + i16 + r + 8 * half;
    const size_t idx  = rowg * FD + n0 + m;
    out[idx +  0] = 0.5f * acc0[r] * q[idx +  0];
    out[idx + 16] = 0.5f * acc1[r] * q[idx + 16];
  }
}

// ---------------------------------------------------------------------------
extern "C" void kernel_launch(void* const* d_in, const int* in_sizes, int n_in,
                              void* d_out, int out_size, void* d_ws, size_t ws_size,
                              hipStream_t stream) {
  const float* positions = (const float*)d_in[0];  // [SYS, NA, 3]
  const float* features  = (const float*)d_in[1];  // [SN, FD]
  const float* W         = (const float*)d_in[2];  // [FD, FD]
  const float* bias      = (const float*)d_in[3];  // [FD]
  float* out = (float*)d_out;                      // [SN, FD]
  float* q   = (float*)d_ws;                       // [SN, FD] scratch (16 MB)

  charges_gemm_k<<<SN / 16, 128, 0, stream>>>(features, W, bias, q);
  potential_gemm_k<<<SYS * (NA / 16), 256, 0, stream>>>(positions, q, out);
}